// CodeBook_29437705846896
// MI455X (gfx1250) — compile-verified
//
#include <hip/hip_runtime.h>

typedef __attribute__((ext_vector_type(2))) float v2f;
typedef __attribute__((ext_vector_type(8))) float v8f;

#define D         256        // embedding dim (K)
#define NCODE     1024       // codebook entries
#define TOK_PER_B 16384      // 16*32*32 tokens per batch element
#define NTOK      131072     // 8 * 16384
#define TOK_PER_BLOCK 128    // 8 waves * 16-token M tile
#define LDS_ROW   258        // 256 + 2 floats pad (8B) -> conflict-free ds_load_b64

// ---------------- kernel 1: codebook squared norms -> d_ws ----------------
__global__ void cnorm_kernel(const float* __restrict__ cb, float* __restrict__ cn) {
    int e = blockIdx.x * blockDim.x + threadIdx.x;
    if (e >= NCODE) return;
    const float4* row = (const float4*)(cb + (size_t)e * D);
    float s = 0.f;
#pragma unroll 4
    for (int i = 0; i < D / 4; ++i) {
        float4 v = row[i];
        s += v.x * v.x + v.y * v.y + v.z * v.z + v.w * v.w;
    }
    cn[e] = s;
}

// ---------------- kernel 2: fp32 WMMA distance GEMM + argmin ----------------
// block = 256 threads = 8 waves; wave w handles tokens [blk*128 + 16w, +16)
// and scans all 1024 codebook entries (64 N-tiles of 16).
__global__ __launch_bounds__(256)
void argmin_kernel(const float* __restrict__ z, const float* __restrict__ cb,
                   const float* __restrict__ cn, int* __restrict__ idx_out,
                   float* __restrict__ idxf_out) {
    __shared__ float zt[TOK_PER_BLOCK * LDS_ROW];  // [token][channel], padded

    const int tid  = threadIdx.x;
    const int lane = tid & 31;
    const int wave = tid >> 5;
    const int tok0 = blockIdx.x * TOK_PER_BLOCK;   // same b for whole block (16384 % 128 == 0)
    const int b    = tok0 / TOK_PER_B;
    const int thw0 = tok0 % TOK_PER_B;
    const float* zb = z + (size_t)b * D * TOK_PER_B + thw0;

    // Stage Z tile: global [c][tok] (coalesced 512B rows) -> LDS [tok][c]
#pragma unroll 4
    for (int i = 0; i < (TOK_PER_BLOCK * D) / 256; ++i) {
        int li  = i * 256 + tid;
        int c   = li >> 7;      // 0..255
        int tok = li & 127;     // 0..127
        zt[tok * LDS_ROW + c] = zb[(size_t)c * TOK_PER_B + tok];
    }
    __syncthreads();

    // A-matrix 16x4 f32 layout: lanes 0-15 hold M=0..15 (K=k0 in vgpr0, k1 in vgpr1),
    // lanes 16-31 hold M=0..15 (K=k2,k3). -> per lane: 2 consecutive channels of row M.
    const int m0    = wave * 16;
    const int mrow  = lane & 15;          // M row (A) / N col (B) this lane serves
    const int khalf = lane >> 4;          // 0 -> channels k,k+1 ; 1 -> k+2,k+3
    const float* zrow = zt + (m0 + mrow) * LDS_ROW + 2 * khalf;

    float bestVal[8];
    int   bestIdx[8];
#pragma unroll
    for (int r = 0; r < 8; ++r) { bestVal[r] = 3.4e38f; bestIdx[r] = 0; }

    for (int nt = 0; nt < NCODE / 16; ++nt) {
        const int n0 = nt * 16;
        // B-matrix 4x16 f32: lane holds entry (n0 + mrow), 2 consecutive channels.
        const float* crow = cb + (size_t)(n0 + mrow) * D + 2 * khalf;
        v8f acc = {};
#pragma unroll 8
        for (int kk = 0; kk < D; kk += 4) {
            v2f a  = *(const v2f*)(zrow + kk);   // ds_load_b64 from LDS
            v2f bb = *(const v2f*)(crow + kk);   // global_load_b64 (L2-resident codebook)
            acc = __builtin_amdgcn_wmma_f32_16x16x4_f32(
                false, a, false, bb, (short)0, acc, false, false);
        }
        // score = ||e||^2 - 2*(z.e)   (||z||^2 constant per row -> dropped)
        const float cnv = cn[n0 + mrow];
        const int   n   = n0 + mrow;     // column this lane holds (same for all 8 rows)
#pragma unroll
        for (int r = 0; r < 8; ++r) {
            float s = __builtin_fmaf(-2.0f, acc[r], cnv);
            if (s < bestVal[r]) { bestVal[r] = s; bestIdx[r] = n; }
        }
    }

    // C/D layout: vgpr r = row M=r (lanes 0-15) and M=r+8 (lanes 16-31).
    // Reduce (min,idx) across the 16 lanes of each half; xor masks 1,2,4,8 stay in-half.
#pragma unroll
    for (int r = 0; r < 8; ++r) {
        float v  = bestVal[r];
        int   ix = bestIdx[r];
#pragma unroll
        for (int m = 1; m <= 8; m <<= 1) {
            float ov = __shfl_xor(v, m, 32);
            int   oi = __shfl_xor(ix, m, 32);
            if (ov < v || (ov == v && oi < ix)) { v = ov; ix = oi; }  // argmin picks first
        }
        bestVal[r] = v; bestIdx[r] = ix;
    }
    if ((lane & 15) == 0) {
        const int half = lane >> 4;
#pragma unroll
        for (int r = 0; r < 8; ++r) {
            int g = tok0 + m0 + r + 8 * half;   // global token id
            idx_out[g]  = bestIdx[r];
            idxf_out[g] = (float)bestIdx[r];
        }
    }
}

// ---------------- kernel 3: gather codebook rows -> quantized & st ----------------
__global__ void gather_kernel(const float* __restrict__ cb, const int* __restrict__ idx,
                              float* __restrict__ qout, float* __restrict__ stout) {
    size_t e = (size_t)blockIdx.x * blockDim.x + threadIdx.x;
    const size_t total = (size_t)NTOK * D;      // 33,554,432 elements
    if (e >= total) return;
    const size_t per_b = (size_t)D * TOK_PER_B;
    size_t b   = e / per_b;
    size_t rem = e - b * per_b;
    size_t c   = rem / TOK_PER_B;
    size_t thw = rem - c * TOK_PER_B;
    int en = idx[b * TOK_PER_B + thw];          // consecutive lanes -> consecutive thw
    float v = cb[(size_t)en * D + c];           // codebook gather, L2-resident
    qout[e]  = v;                               // coalesced
    stout[e] = v;                               // st == quantized numerically
}

// ---------------- launcher ----------------
extern "C" void kernel_launch(void* const* d_in, const int* in_sizes, int n_in,
                              void* d_out, int out_size, void* d_ws, size_t ws_size,
                              hipStream_t stream) {
    const float* z  = (const float*)d_in[0];   // [8, 256, 16, 32, 32]
    const float* cb = (const float*)d_in[1];   // [1024, 256]

    float* qout  = (float*)d_out;                       // quantized: 33,554,432 f32
    float* stout = qout + (size_t)NTOK * D;             // st:        33,554,432 f32
    float* idxf  = stout + (size_t)NTOK * D;            // indices:   131,072 (as float)

    float* cn  = (float*)d_ws;                          // 1024 f32 codebook norms
    int*   idx = (int*)((char*)d_ws + NCODE * sizeof(float)); // 131,072 i32 scratch

    cnorm_kernel<<<(NCODE + 255) / 256, 256, 0, stream>>>(cb, cn);
    argmin_kernel<<<NTOK / TOK_PER_BLOCK, 256, 0, stream>>>(z, cb, cn, idx, idxf);
    gather_kernel<<<(int)(((size_t)NTOK * D) / 256), 256, 0, stream>>>(cb, idx, qout, stout);
}